// LocEncoder_41188736369203
// MI455X (gfx1250) — compile-verified
//
#include <hip/hip_runtime.h>

typedef float v2f __attribute__((ext_vector_type(2)));
typedef float v8f __attribute__((ext_vector_type(8)));

#define F_DIM 64
#define OUT_DIM 7

// ---------------- zero workspace (agg + cnt) ----------------
__global__ void zero_ws_kernel(float* __restrict__ ws, int n) {
    int i = blockIdx.x * blockDim.x + threadIdx.x;
    if (i < n) ws[i] = 0.0f;
}

// ---------------- edge scatter: 16 lanes per edge ----------------
// lane q of an edge loads x[src][4q..4q+3] as float4 (coalesced 256B per edge)
// and atomically accumulates into agg[dst]. Lane 0 bumps the degree count.
// x (25.6MB) and agg (25.6MB) are L2-resident on a 192MB L2, so this is
// L2-atomic-throughput bound; HBM traffic is essentially edge_index + one
// pass of x/agg.
__global__ __launch_bounds__(256) void edge_scatter_kernel(
    const float* __restrict__ x, const long long* __restrict__ ei,
    float* __restrict__ agg, float* __restrict__ cnt, int E) {
    int t = blockIdx.x * blockDim.x + threadIdx.x;
    int e = t >> 4;
    int q = t & 15;
    if (e >= E) return;
    long long src = ei[e];                 // edge_index[0][e]
    long long dst = ei[(long long)E + e];  // edge_index[1][e]
    const float4 v = *(const float4*)(x + src * F_DIM + q * 4);
    float* a = agg + dst * F_DIM + q * 4;
    atomicAdd(a + 0, v.x);
    atomicAdd(a + 1, v.y);
    atomicAdd(a + 2, v.z);
    atomicAdd(a + 3, v.w);
    if (q == 0) atomicAdd(cnt + dst, 1.0f);
}

// ---------------- fused node GEMM via V_WMMA_F32_16X16X4_F32 ----------------
// out[m, 0:7] = relu( [mean(m) | x(m)] (1x128)  @  [W_l ; W_r] (128x16pad) + b )
// One wave per 16-row tile; K=128 in 32 steps of the 16x16x4 f32 WMMA.
//
// A fragment (16x4 f32, ISA 7.12.2): lanes 0-15 hold {K,K+1} of row M=lane,
// lanes 16-31 hold {K+2,K+3} of row M=lane-16.
// B fragment (4x16 f32): v0/v1 lanes 0-15 = rows K,K+1 (N=lane),
// lanes 16-31 = rows K+2,K+3 (N=lane-16).
// C/D (16x16 f32): VGPR r: lanes 0-15 -> M=r, lanes 16-31 -> M=r+8, N=lane%16.
__global__ __launch_bounds__(256) void node_gemm_kernel(
    const float* __restrict__ x, const float* __restrict__ agg,
    const float* __restrict__ cnt, const float* __restrict__ W_l,
    const float* __restrict__ b_l, const float* __restrict__ W_r,
    float* __restrict__ out, int N) {
    __shared__ float ldsW[128 * 16];  // [K=128][Npad=16], 8KB

    // Cooperative fill of concatenated, zero-padded weight matrix.
    for (int i = threadIdx.x; i < 128 * 16; i += 256) {
        int kk = i >> 4;
        int n  = i & 15;
        float w = 0.0f;
        if (n < OUT_DIM)
            w = (kk < F_DIM) ? W_l[kk * OUT_DIM + n]
                             : W_r[(kk - F_DIM) * OUT_DIM + n];
        ldsW[i] = w;
    }
    __syncthreads();

    const int lane = threadIdx.x & 31;
    const int wv   = threadIdx.x >> 5;
    const int m0   = blockIdx.x * 128 + wv * 16;
    if (m0 >= N) return;  // uniform per wave; N % 16 == 0 so live tiles are full

    const int half = lane >> 4;   // 0 -> K+0/K+1, 1 -> K+2/K+3
    const int col  = lane & 15;
    const int row  = m0 + col;    // A-matrix row owned by this lane
    const int koff = half * 2;

    const float inv = 1.0f / fmaxf(cnt[row], 1.0f);
    const float* aggRow = agg + (size_t)row * F_DIM;
    const float* xRow   = x   + (size_t)row * F_DIM;

    v8f acc = {};
#pragma unroll
    for (int k = 0; k < 128; k += 4) {
        const int f = k + koff;  // even -> 8B-aligned float2 loads
        v2f a;
        if (f < F_DIM) {  // mean half: agg * (1/max(cnt,1))
            float2 t = *(const float2*)(aggRow + f);
            a.x = t.x * inv;
            a.y = t.y * inv;
        } else {          // x half
            float2 t = *(const float2*)(xRow + (f - F_DIM));
            a.x = t.x;
            a.y = t.y;
        }
        v2f b;
        b.x = ldsW[f * 16 + col];
        b.y = ldsW[(f + 1) * 16 + col];
        // (neg_a, A, neg_b, B, c_mod, C, reuse_a, reuse_b)
        acc = __builtin_amdgcn_wmma_f32_16x16x4_f32(
            false, a, false, b, (short)0, acc, false, false);
    }

    const float bias = (col < OUT_DIM) ? b_l[col] : 0.0f;
#pragma unroll
    for (int r = 0; r < 8; ++r) {
        const int orow = m0 + r + 8 * half;
        if (col < OUT_DIM) {
            out[(size_t)orow * OUT_DIM + col] = fmaxf(acc[r] + bias, 0.0f);
        }
    }
}

extern "C" void kernel_launch(void* const* d_in, const int* in_sizes, int n_in,
                              void* d_out, int out_size, void* d_ws, size_t ws_size,
                              hipStream_t stream) {
    const float*     x   = (const float*)d_in[0];
    const long long* ei  = (const long long*)d_in[1];
    const float*     W_l = (const float*)d_in[2];
    const float*     b_l = (const float*)d_in[3];
    const float*     W_r = (const float*)d_in[4];
    float* out = (float*)d_out;

    const int N = in_sizes[0] / F_DIM;  // 100000
    const int E = in_sizes[1] / 2;      // 1600000

    float* agg = (float*)d_ws;                    // N*64 f32
    float* cnt = agg + (size_t)N * F_DIM;         // N f32

    const int nz = N * (F_DIM + 1);
    zero_ws_kernel<<<(nz + 255) / 256, 256, 0, stream>>>((float*)d_ws, nz);

    const long long tEdge = (long long)E * 16;
    edge_scatter_kernel<<<(unsigned)((tEdge + 255) / 256), 256, 0, stream>>>(
        x, ei, agg, cnt, E);

    const int nBlocks = (N + 127) / 128;  // 128 rows (8 wave-tiles) per block
    node_gemm_kernel<<<nBlocks, 256, 0, stream>>>(x, agg, cnt, W_l, b_l, W_r,
                                                  out, N);
}